// Extraction_spatial_features_78323023610545
// MI455X (gfx1250) — compile-verified
//
#include <hip/hip_runtime.h>

// Problem constants (match reference)
#define BT_    192          // B*T
#define N_     1000         // logical node count
#define NP_    1008         // padded node stride in workspace (multiple of 16)
#define K_     16           // neighbors per node
#define C1_    32
#define C2_    16
#define NK_    16000        // N_*K_ (flat neighbor table size, logical)
#define NTILES 63           // ceil(N_/16)

typedef __attribute__((ext_vector_type(2))) float v2f;
typedef __attribute__((ext_vector_type(4))) float v4f;
typedef __attribute__((ext_vector_type(8))) float v8f;
typedef __attribute__((ext_vector_type(2))) int   v2i;
typedef __attribute__((ext_vector_type(4))) int   v4i;

// D(16x16,f32) = A(16x4,f32) * B(4x16,f32) + C  -- fp32 matrix pipe, fp32 accum.
__device__ __forceinline__ v8f wmma_f32_4(v2f a, v2f b, v8f c) {
    return __builtin_amdgcn_wmma_f32_16x16x4_f32(
        /*neg_a=*/false, a, /*neg_b=*/false, b,
        /*c_mod=*/(short)0, c, /*reuse_a=*/false, /*reuse_b=*/false);
}

// ---------------------------------------------------------------------------
// Layer 1: h1[bt,c,n] = relu( sum_k x[bt, nb[n,k]] * W1[c,k] + b1[c] )
// One wave per 16-node tile; two 16-channel B tiles; 4 WMMA k-steps each.
// x[bt] is 4KB and L2-hot; global gathers are fine here (6% of FLOPs).
// ---------------------------------------------------------------------------
__global__ __launch_bounds__(256)
void k1_layer1(const float* __restrict__ x, const int* __restrict__ nb,
               const float* __restrict__ W1, const float* __restrict__ b1,
               float* __restrict__ h1) {
    const int lane   = threadIdx.x & 31;
    const int wave   = threadIdx.x >> 5;
    const int tileId = blockIdx.x * 8 + wave;
    if (tileId >= BT_ * NTILES) return;           // wave-uniform
    const int bt   = tileId / NTILES;
    const int base = (tileId % NTILES) * 16;

    const int mrow  = lane & 15;                  // A row / B col / D col
    const int hi    = lane >> 4;
    const int khalf = hi * 2;                     // k-pair base within k-step
    const int node  = min(base + mrow, N_ - 1);   // clamp tail gathers
    const float* xb = x + bt * N_;

    for (int ct = 0; ct < 2; ++ct) {
        v8f acc = {};
#pragma unroll
        for (int s = 0; s < 4; ++s) {
            const int k0 = s * 4 + khalf;
            const v2i idx = *(const v2i*)&nb[node * K_ + k0];   // even, 8B aligned
            v2f a, b;
            a.x = xb[idx.x];
            a.y = xb[idx.y];
            b = *(const v2f*)&W1[(ct * 16 + mrow) * K_ + k0];   // B[k][c]=W1[c][k]
            acc = wmma_f32_4(a, b, acc);
        }
        const int   ch   = ct * 16 + mrow;        // D col -> channel
        const float bias = b1[ch];
        v4f lo, hiv;
#pragma unroll
        for (int j = 0; j < 4; ++j) {
            float v0 = acc[j]     + bias;  lo[j]  = v0 > 0.f ? v0 : 0.f;
            float v1 = acc[j + 4] + bias;  hiv[j] = v1 > 0.f ? v1 : 0.f;
        }
        float* p = h1 + (size_t)(bt * C1_ + ch) * NP_ + base + 8 * hi;
        *(v4f*)p       = lo;                      // 16B aligned
        *(v4f*)(p + 4) = hiv;
    }
}

// ---------------------------------------------------------------------------
// Layer 2: dominant GEMM (192k x 512) x (512 x 16). Row n, feature f:
//   flat = n*512 + f, c = flat/16000, rem = flat%16000, A = h1[bt, c, nb[rem]]
// (faithful raw-reshape C/N axis mixing; flat math uses LOGICAL N).
// CDNA5 320KB-LDS design: the whole per-bt slice of h1 (129KB), the neighbor
// table (64KB) and W2 (32KB) live in LDS -> the 128-step WMMA loop does ZERO
// global traffic. Two blocks per bt (8 waves x 4 tiles each).
// ---------------------------------------------------------------------------
__global__ __launch_bounds__(256)
void k2_layer2(const float* __restrict__ h1, const int* __restrict__ nb,
               const float* __restrict__ W2, const float* __restrict__ b2,
               const float* __restrict__ gamma, const float* __restrict__ beta,
               float* __restrict__ h2) {
    __shared__ float h1s[C1_ * NP_];              // 129,024 B
    __shared__ int   nbs[NK_];                    //  64,000 B
    __shared__ float w2s[C2_ * 512];              //  32,768 B   (total ~226 KB)

    const int bt     = blockIdx.x >> 1;
    const int halfId = blockIdx.x & 1;

    // Stage per-bt activations + shared tables with b128 streams.
    const float* h1b = h1 + (size_t)bt * C1_ * NP_;
    for (int i = threadIdx.x; i < C1_ * NP_ / 4; i += 256)
        ((v4f*)h1s)[i] = ((const v4f*)h1b)[i];
    for (int i = threadIdx.x; i < NK_ / 4; i += 256)
        ((v4i*)nbs)[i] = ((const v4i*)nb)[i];
    for (int i = threadIdx.x; i < C2_ * 512 / 4; i += 256)
        ((v4f*)w2s)[i] = ((const v4f*)W2)[i];
    __syncthreads();

    const int lane  = threadIdx.x & 31;
    const int wave  = threadIdx.x >> 5;
    const int mrow  = lane & 15;
    const int hi    = lane >> 4;
    const int khalf = hi * 2;

    const float scale = gamma[mrow] * rsqrtf(1.0f + 1e-5f);
    const float bias  = b2[mrow];
    const float shift = beta[mrow];

    for (int t = 0; t < 4; ++t) {
        const int tile = halfId * 32 + wave * 4 + t;    // wave-uniform
        if (tile >= NTILES) continue;                   // only tile 63 skipped
        const int base = tile * 16;
        const int node = min(base + mrow, N_ - 1);
        const unsigned fbase = (unsigned)node * 512u;

        v8f acc = {};
#pragma unroll 8
        for (int s = 0; s < 128; ++s) {
            const int k0 = s * 4 + khalf;
            // f0 always even -> f0,f0+1 share the same c slice, r is even.
            const unsigned f0 = fbase + (unsigned)k0;
            const unsigned c  = f0 / (unsigned)NK_;
            const unsigned r  = f0 - c * (unsigned)NK_;     // == nn*K + kk
            const v2i idx = *(const v2i*)&nbs[r];           // ds_load_b64
            v2f a, b;
            a.x = h1s[c * NP_ + idx.x];                     // LDS gathers
            a.y = h1s[c * NP_ + idx.y];
            b = *(const v2f*)&w2s[mrow * 512 + k0];         // B[k][c] = W2[c][k]
            acc = wmma_f32_4(a, b, acc);
        }
        v4f lo, hiv;
#pragma unroll
        for (int j = 0; j < 4; ++j) {
            float v0 = (acc[j]     + bias) * scale + shift;  lo[j]  = v0 > 0.f ? v0 : 0.f;
            float v1 = (acc[j + 4] + bias) * scale + shift;  hiv[j] = v1 > 0.f ? v1 : 0.f;
        }
        float* p = h2 + (size_t)(bt * C2_ + mrow) * NP_ + base + 8 * hi;
        *(v4f*)p       = lo;
        *(v4f*)(p + 4) = hiv;
    }
}

// ---------------------------------------------------------------------------
// Layer 3: out[bt,n] = relu( sum_{f<256} h2[bt, c(f), nb[rem(f)]] * W3[f] + b3 )
// Output dim is 1 -> WMMA would waste 15/16 columns; LDS-resident 256-MAC dot.
// Two blocks per bt, 500 nodes each, all gathers from LDS.
// ---------------------------------------------------------------------------
__global__ __launch_bounds__(256)
void k3_layer3(const float* __restrict__ h2, const int* __restrict__ nb,
               const float* __restrict__ W3, const float* __restrict__ b3,
               float* __restrict__ out) {
    __shared__ float h2s[C2_ * NP_];              // 64,512 B
    __shared__ int   nbs[NK_];                    // 64,000 B
    __shared__ float w3s[256];                    //  1,024 B   (total ~130 KB)

    const int bt     = blockIdx.x >> 1;
    const int halfId = blockIdx.x & 1;

    const float* h2b = h2 + (size_t)bt * C2_ * NP_;
    for (int i = threadIdx.x; i < C2_ * NP_ / 4; i += 256)
        ((v4f*)h2s)[i] = ((const v4f*)h2b)[i];
    for (int i = threadIdx.x; i < NK_ / 4; i += 256)
        ((v4i*)nbs)[i] = ((const v4i*)nb)[i];
    if (threadIdx.x < 256) w3s[threadIdx.x] = W3[threadIdx.x];
    __syncthreads();

    const float b3v = b3[0];
    for (int i = 0; i < 2; ++i) {
        const int off = i * 256 + threadIdx.x;          // 0..499 covered
        if (off >= 500) break;
        const int n = halfId * 500 + off;
        float acc = b3v;
        const unsigned fbase = (unsigned)n * 256u;
#pragma unroll 8
        for (int f = 0; f < 256; f += 2) {
            const unsigned fl = fbase + (unsigned)f;    // even -> paired index
            const unsigned c  = fl / (unsigned)NK_;
            const unsigned r  = fl - c * (unsigned)NK_;
            const v2i idx = *(const v2i*)&nbs[r];
            acc = fmaf(h2s[c * NP_ + idx.x], w3s[f],     acc);
            acc = fmaf(h2s[c * NP_ + idx.y], w3s[f + 1], acc);
        }
        out[bt * N_ + n] = acc > 0.f ? acc : 0.f;
    }
}

// ---------------------------------------------------------------------------
extern "C" void kernel_launch(void* const* d_in, const int* in_sizes, int n_in,
                              void* d_out, int out_size, void* d_ws, size_t ws_size,
                              hipStream_t stream) {
    const float* x     = (const float*)d_in[0];
    const int*   nb    = (const int*)  d_in[1];
    const float* W1    = (const float*)d_in[2];
    const float* b1    = (const float*)d_in[3];
    const float* W2    = (const float*)d_in[4];
    const float* b2    = (const float*)d_in[5];
    const float* W3    = (const float*)d_in[6];
    const float* b3    = (const float*)d_in[7];
    const float* gamma = (const float*)d_in[8];
    const float* beta  = (const float*)d_in[9];

    float* h1  = (float*)d_ws;                         // (BT, C1, NP) ~24.8 MB
    float* h2  = h1 + (size_t)BT_ * C1_ * NP_;         // (BT, C2, NP) ~12.4 MB
    float* out = (float*)d_out;

    const int tiles  = BT_ * NTILES;                   // 12096 waves for k1
    const int blocks = (tiles + 7) / 8;

    k1_layer1<<<blocks,  256, 0, stream>>>(x, nb, W1, b1, h1);
    k2_layer2<<<BT_ * 2, 256, 0, stream>>>(h1, nb, W2, b2, gamma, beta, h2);
    k3_layer3<<<BT_ * 2, 256, 0, stream>>>(h2, nb, W3, b3, out);
}